// Critic_6760278524119
// MI455X (gfx1250) — compile-verified
//
#include <hip/hip_runtime.h>

#define HH   4      // heads
#define CH   32     // channels per head
#define HID  128    // HH*CH

typedef __attribute__((ext_vector_type(2))) float v2f;
typedef __attribute__((ext_vector_type(8))) float v8f;

// ---------- float <-> orderable unsigned (for atomic float max) ----------
__device__ __forceinline__ unsigned fenc(float f) {
  int b = __float_as_int(f);
  return (b >= 0) ? ((unsigned)b | 0x80000000u) : (unsigned)(~b);
}
__device__ __forceinline__ float fdec(unsigned u) {
  int b = (u & 0x80000000u) ? (int)(u & 0x7fffffffu) : ~(int)u;
  return __int_as_float(b);
}

// ---------- W repack into WMMA B-fragment order ----------
// Wp flat index: ((g*2 + half)*2 + cb)*128 + l16*8 + j
//   g = k-group (4 k's), half = lane>>4, cb = 64-col strip, l16 = lane&15,
//   j = {pair p = j>>1 -> col offset 16p, parity q = j&1 -> row 4g+2*half+q}
// so each lane's 8 fragment floats for one k-step are contiguous (2x b128).
__global__ void repack_W(const float* __restrict__ W, float* __restrict__ Wp, int K) {
  int t = blockIdx.x * blockDim.x + threadIdx.x;
  if (t >= K * HID) return;
  int j    = t & 7;
  int l16  = (t >> 3) & 15;
  int cb   = (t >> 7) & 1;
  int half = (t >> 8) & 1;
  int g    = t >> 9;
  int q = j & 1, p = j >> 1;
  Wp[t] = W[(size_t)(4 * g + 2 * half + q) * HID + cb * 64 + 16 * p + l16];
}

// ---------- WMMA GEMM: Y[N x 128] = X[N x K] @ Wp(repacked) ----------
// One wave computes a 16x64 strip: 4 chained v_wmma_f32_16x16x4_f32 per k-step.
// Per k-step loads: 1x global_load_b64 (A) + 2x global_load_b128 (B frags).
// N is a multiple of 16 here, so the only guard is wave-uniform (EXEC all-1s
// through every WMMA, as the ISA requires).
template <int K>
__global__ void gemm_wmma(const float* __restrict__ X, const float* __restrict__ Wp,
                          float* __restrict__ Y, int N) {
  const int wave = threadIdx.x >> 5;              // 4 waves per block
  const int lane = threadIdx.x & 31;
  const int rowTile = (blockIdx.x * 4 + wave) * 16;
  if (rowTile >= N) return;                       // wave-uniform guard
  const int colBase = blockIdx.y * 64;            // gridDim.y == 2
  const int half = lane >> 4;
  const int l16  = lane & 15;

  const float* xrow = X + (size_t)(rowTile + l16) * K + 2 * half;
  const float* wp   = Wp + (size_t)(half * 2 + blockIdx.y) * 128 + l16 * 8;

  v8f acc0 = {}, acc1 = {}, acc2 = {}, acc3 = {};
#pragma unroll
  for (int g = 0; g < K / 4; ++g) {
    const v2f    a  = *(const v2f*)   (xrow + 4 * g);        // global_load_b64
    const float4 lo = *(const float4*)(wp + g * 512);        // global_load_b128
    const float4 hi = *(const float4*)(wp + g * 512 + 4);    // global_load_b128
    v2f b0 = {lo.x, lo.y};
    v2f b1 = {lo.z, lo.w};
    v2f b2 = {hi.x, hi.y};
    v2f b3 = {hi.z, hi.w};
    acc0 = __builtin_amdgcn_wmma_f32_16x16x4_f32(false, a, false, b0, (short)0, acc0, false, false);
    acc1 = __builtin_amdgcn_wmma_f32_16x16x4_f32(false, a, false, b1, (short)0, acc1, false, false);
    acc2 = __builtin_amdgcn_wmma_f32_16x16x4_f32(false, a, false, b2, (short)0, acc2, false, false);
    acc3 = __builtin_amdgcn_wmma_f32_16x16x4_f32(false, a, false, b3, (short)0, acc3, false, false);
  }
#pragma unroll
  for (int j = 0; j < 8; ++j) {
    const int m = rowTile + j + 8 * half;
    float* yrow = Y + (size_t)m * HID + colBase + l16;
    yrow[0]  = acc0[j];
    yrow[16] = acc1[j];
    yrow[32] = acc2[j];
    yrow[48] = acc3[j];
  }
}

// ---------- per-node attention coefficients ----------
__global__ void attn_coeff(const float* __restrict__ xh,
                           const float* __restrict__ att_src,
                           const float* __restrict__ att_dst,
                           float* __restrict__ asrc, float* __restrict__ adst, int N) {
  int t = blockIdx.x * blockDim.x + threadIdx.x;
  if (t >= N * HH) return;
  const int h = t & 3;
  const float4* row = (const float4*)(xh + (size_t)(t >> 2) * HID + h * CH);
  const float4* as4 = (const float4*)(att_src + h * CH);
  const float4* ad4 = (const float4*)(att_dst + h * CH);
  float ss = 0.f, sd = 0.f;
#pragma unroll
  for (int c = 0; c < CH / 4; ++c) {
    float4 v = row[c], a = as4[c], d = ad4[c];
    ss += v.x * a.x + v.y * a.y + v.z * a.z + v.w * a.w;
    sd += v.x * d.x + v.y * d.y + v.z * d.z + v.w * d.w;
  }
  asrc[t] = ss;
  adst[t] = sd;
}

// ---------- edge pass 1: leaky-relu logits + segment max ----------
__global__ void edge_alpha_max(const int* __restrict__ ei, int E, int N,
                               const float* __restrict__ asrc, const float* __restrict__ adst,
                               float* __restrict__ ebuf, unsigned* __restrict__ amax) {
  int t = blockIdx.x * blockDim.x + threadIdx.x;
  int total = (E + N) * HH;
  if (t >= total) return;
  int eh = t >> 2, h = t & 3;
  int s, d;
  if (eh < E) { s = ei[eh]; d = ei[E + eh]; } else { s = d = eh - E; }
  float a = asrc[s * HH + h] + adst[d * HH + h];
  a = (a > 0.f) ? a : 0.2f * a;             // leaky relu, slope 0.2
  ebuf[t] = a;
  atomicMax(&amax[d * HH + h], fenc(a));
}

// ---------- edge pass 2: exp(alpha - max) + segment sum ----------
__global__ void edge_exp_sum(const int* __restrict__ ei, int E, int N,
                             float* __restrict__ ebuf, const unsigned* __restrict__ amax,
                             float* __restrict__ denom) {
  int t = blockIdx.x * blockDim.x + threadIdx.x;
  int total = (E + N) * HH;
  if (t >= total) return;
  int eh = t >> 2, h = t & 3;
  int d = (eh < E) ? ei[E + eh] : (eh - E);
  float e = __expf(ebuf[t] - fdec(amax[d * HH + h]));
  ebuf[t] = e;
  atomicAdd(&denom[d * HH + h], e);
}

// ---------- edge pass 3: weighted message scatter ----------
__global__ void edge_scatter(const int* __restrict__ ei, int E, int N,
                             const float* __restrict__ ebuf, const float* __restrict__ denom,
                             const float* __restrict__ xh, float* __restrict__ agg) {
  int t = blockIdx.x * blockDim.x + threadIdx.x;
  int total = (E + N) * HH;
  if (t >= total) return;
  int eh = t >> 2, h = t & 3;
  int s, d;
  if (eh < E) { s = ei[eh]; d = ei[E + eh]; } else { s = d = eh - E; }
  float w = ebuf[t] / denom[d * HH + h];
  const float4* xrow = (const float4*)(xh + (size_t)s * HID + h * CH);
  float* orow = agg + (size_t)d * HID + h * CH;
#pragma unroll
  for (int c = 0; c < CH / 4; ++c) {
    float4 v = xrow[c];
    atomicAdd(&orow[4 * c + 0], v.x * w);
    atomicAdd(&orow[4 * c + 1], v.y * w);
    atomicAdd(&orow[4 * c + 2], v.z * w);
    atomicAdd(&orow[4 * c + 3], v.w * w);
  }
}

// ---------- bias + relu ----------
__global__ void finalize_relu(const float* __restrict__ agg, const float* __restrict__ bias,
                              float* __restrict__ hout, int N) {
  int t = blockIdx.x * blockDim.x + threadIdx.x;
  if (t >= N * HID) return;
  hout[t] = fmaxf(agg[t] + bias[t & (HID - 1)], 0.f);
}

// ---------- column sum (mean computed later by scaling) ----------
__global__ void colsum(const float* __restrict__ h, int N, float* __restrict__ acc) {
  const int c = threadIdx.x;          // 128 threads
  float s = 0.f;
  for (int n = blockIdx.x; n < N; n += gridDim.x) s += h[(size_t)n * HID + c];
  atomicAdd(&acc[c], s);
}

// ---------- fused projection-mean + MLP head ----------
__global__ void head_kernel(const float* __restrict__ hmP, const float* __restrict__ hmV,
                            const float* __restrict__ pWp, const float* __restrict__ pbp,
                            const float* __restrict__ pWv, const float* __restrict__ pbv,
                            const float* __restrict__ W0, const float* __restrict__ b0,
                            const float* __restrict__ W1, const float* __restrict__ b1,
                            const float* __restrict__ W2, const float* __restrict__ b2,
                            float invNp, float invNv, float* __restrict__ out) {
  __shared__ float fused[2 * HID];
  __shared__ float h1[HID];
  __shared__ float h2[HID / 2];
  const int t = threadIdx.x;          // 256 threads
  {
    float s;
    if (t < HID) {
      s = pbp[t];
      for (int k = 0; k < HID; ++k) s += hmP[k] * invNp * pWp[k * HID + t];
    } else {
      int c = t - HID;
      s = pbv[c];
      for (int k = 0; k < HID; ++k) s += hmV[k] * invNv * pWv[k * HID + c];
    }
    fused[t] = s;
  }
  __syncthreads();
  if (t < HID) {
    float s = b0[t];
    for (int i = 0; i < 2 * HID; ++i) s += fused[i] * W0[i * HID + t];
    h1[t] = fmaxf(s, 0.f);
  }
  __syncthreads();
  if (t < HID / 2) {
    float s = b1[t];
    for (int i = 0; i < HID; ++i) s += h1[i] * W1[i * (HID / 2) + t];
    h2[t] = fmaxf(s, 0.f);
  }
  __syncthreads();
  if (t == 0) {
    float s = b2[0];
    for (int i = 0; i < HID / 2; ++i) s += h2[i] * W2[i];
    out[0] = s;
  }
}

// ---------- host-side encoder driver ----------
static void run_encoder(const float* feats, int in_dim, const int* ei, int E, int N,
                        const float* Ws[3], const float* attd[3], const float* atts[3],
                        const float* bias[3],
                        float* hbuf, float* xh, float* agg, float* asrc, float* adst,
                        unsigned* amax, float* denom, float* ebuf, float* wpack,
                        float* hsum, hipStream_t stream) {
  for (int l = 0; l < 3; ++l) {
    const int K = (l == 0) ? in_dim : HID;
    const float* X = (l == 0) ? feats : hbuf;

    repack_W<<<(K * HID + 255) / 256, 256, 0, stream>>>(Ws[l], wpack, K);

    dim3 grid((N + 63) / 64, 2);                 // 4 row-tiles/block, 64-col strips
    if (K == 32)       gemm_wmma<32 ><<<grid, 128, 0, stream>>>(X, wpack, xh, N);
    else if (K == 64)  gemm_wmma<64 ><<<grid, 128, 0, stream>>>(X, wpack, xh, N);
    else               gemm_wmma<128><<<grid, 128, 0, stream>>>(X, wpack, xh, N);

    int nh = N * HH;
    attn_coeff<<<(nh + 255) / 256, 256, 0, stream>>>(xh, atts[l], attd[l], asrc, adst, N);

    hipMemsetAsync(agg,   0, (size_t)N * HID * sizeof(float), stream);
    hipMemsetAsync(amax,  0, (size_t)N * HH * sizeof(unsigned), stream);
    hipMemsetAsync(denom, 0, (size_t)N * HH * sizeof(float), stream);

    int work = (E + N) * HH;
    int blk  = (work + 255) / 256;
    edge_alpha_max<<<blk, 256, 0, stream>>>(ei, E, N, asrc, adst, ebuf, amax);
    edge_exp_sum  <<<blk, 256, 0, stream>>>(ei, E, N, ebuf, amax, denom);
    edge_scatter  <<<blk, 256, 0, stream>>>(ei, E, N, ebuf, denom, xh, agg);

    int nt = N * HID;
    finalize_relu<<<(nt + 255) / 256, 256, 0, stream>>>(agg, bias[l], hbuf, N);
  }
  hipMemsetAsync(hsum, 0, HID * sizeof(float), stream);
  colsum<<<256, HID, 0, stream>>>(hbuf, N, hsum);
}

extern "C" void kernel_launch(void* const* d_in, const int* in_sizes, int n_in,
                              void* d_out, int out_size, void* d_ws, size_t ws_size,
                              hipStream_t stream) {
  (void)n_in; (void)out_size; (void)ws_size;
  // graph arrays (dict insertion order)
  const float* pfeat = (const float*)d_in[0];
  const int*   pei   = (const int*)  d_in[1];
  const float* vfeat = (const float*)d_in[3];
  const int*   vei   = (const int*)  d_in[4];
  const int Np = in_sizes[0] / 64;
  const int Ep = in_sizes[1] / 2;
  const int Nv = in_sizes[3] / 32;
  const int Ev = in_sizes[4] / 2;

  // params flattened in JAX sorted-key DFS order: head, p, v
  const float* W0 = (const float*)d_in[6];   const float* b0 = (const float*)d_in[7];
  const float* W1 = (const float*)d_in[8];   const float* b1 = (const float*)d_in[9];
  const float* W2 = (const float*)d_in[10];  const float* b2 = (const float*)d_in[11];

  const int PB = 12, VB = 26;   // encoder param bases; conv l at base+4l: W, att_dst, att_src, bias
  const float* pW[3], *pAd[3], *pAs[3], *pBi[3];
  const float* vW[3], *vAd[3], *vAs[3], *vBi[3];
  for (int l = 0; l < 3; ++l) {
    pW[l]  = (const float*)d_in[PB + 4 * l + 0];
    pAd[l] = (const float*)d_in[PB + 4 * l + 1];
    pAs[l] = (const float*)d_in[PB + 4 * l + 2];
    pBi[l] = (const float*)d_in[PB + 4 * l + 3];
    vW[l]  = (const float*)d_in[VB + 4 * l + 0];
    vAd[l] = (const float*)d_in[VB + 4 * l + 1];
    vAs[l] = (const float*)d_in[VB + 4 * l + 2];
    vBi[l] = (const float*)d_in[VB + 4 * l + 3];
  }
  const float* projWp = (const float*)d_in[PB + 12];
  const float* projbp = (const float*)d_in[PB + 13];
  const float* projWv = (const float*)d_in[VB + 12];
  const float* projbv = (const float*)d_in[VB + 13];

  // workspace carve
  const size_t NM = (size_t)((Np > Nv) ? Np : Nv);
  const size_t EM = (size_t)((Ep > Ev) ? Ep : Ev);
  size_t off = 0;
  auto carve = [&](size_t bytes) -> void* {
    void* p = (char*)d_ws + off;
    off += (bytes + 255) & ~(size_t)255;
    return p;
  };
  float*    hbuf  = (float*)   carve(NM * HID * sizeof(float));
  float*    xh    = (float*)   carve(NM * HID * sizeof(float));
  float*    agg   = (float*)   carve(NM * HID * sizeof(float));
  float*    asrc  = (float*)   carve(NM * HH * sizeof(float));
  float*    adst  = (float*)   carve(NM * HH * sizeof(float));
  unsigned* amax  = (unsigned*)carve(NM * HH * sizeof(unsigned));
  float*    denom = (float*)   carve(NM * HH * sizeof(float));
  float*    ebuf  = (float*)   carve((EM + NM) * HH * sizeof(float));
  float*    wpack = (float*)   carve((size_t)HID * HID * sizeof(float));
  float*    hsumP = (float*)   carve(HID * sizeof(float));
  float*    hsumV = (float*)   carve(HID * sizeof(float));

  run_encoder(pfeat, 64, pei, Ep, Np, pW, pAd, pAs, pBi,
              hbuf, xh, agg, asrc, adst, amax, denom, ebuf, wpack, hsumP, stream);
  run_encoder(vfeat, 32, vei, Ev, Nv, vW, vAd, vAs, vBi,
              hbuf, xh, agg, asrc, adst, amax, denom, ebuf, wpack, hsumV, stream);

  head_kernel<<<1, 256, 0, stream>>>(hsumP, hsumV, projWp, projbp, projWv, projbv,
                                     W0, b0, W1, b1, W2, b2,
                                     1.0f / (float)Np, 1.0f / (float)Nv,
                                     (float*)d_out);
}